// ULayer_2379411882437
// MI455X (gfx1250) — compile-verified
//
#include <hip/hip_runtime.h>

// ---------------------------------------------------------------------------
// ULayer: out = (U1 (x) U2 (x) U3) @ psi, psi in [128, 4096] complex (f32).
// Factored tensor contraction, fused single kernel, state in LDS,
// all GEMMs via V_WMMA_F32_16X16X4_F32 chains (full f32 precision).
// ---------------------------------------------------------------------------

typedef float v2f __attribute__((ext_vector_type(2)));
typedef float v8f __attribute__((ext_vector_type(8)));

#define DIM   4096
#define BATCH 128

static __device__ __forceinline__ v8f wmma_f32(v2f a, v2f b, v8f c) {
  // D = A(16x4 f32) x B(4x16 f32) + C(16x16 f32)
  return __builtin_amdgcn_wmma_f32_16x16x4_f32(
      /*neg_a=*/false, a, /*neg_b=*/false, b,
      /*c_mod=*/(short)0, c, /*reuse_a=*/false, /*reuse_b=*/false);
}

// psi linear index = a*256 + b*16 + c  (a,b,c in [0,16))
// STAGE 0 contracts c (gate U3), STAGE 1 contracts b (U2), STAGE 2 contracts a (U1).
// Each stage: out-tile = U @ in-tile, where the tile's "k" axis is the
// contracted axis and "col" enumerates the 16 bystander indices of tile t.
template <int STAGE>
static __device__ __forceinline__ int map_idx(int t, int k, int col) {
  if constexpr (STAGE == 0)      return ((t * 16 + col) << 4) + k;   // rows=(a,b), k=c
  else if constexpr (STAGE == 1) return (t << 8) + (k << 4) + col;   // t=a, k=b, col=c
  else                           return (k << 8) + t * 16 + col;     // k=a, cols=(b,c)
}

template <int STAGE>
static __device__ __forceinline__ void do_stage(const float* __restrict__ G,
                                                float* SR, float* SI,
                                                float* __restrict__ gOutR,
                                                float* __restrict__ gOutI) {
  const int tid    = threadIdx.x;
  const int lane16 = tid & 15;        // N (B/C/D column) / A row
  const int hi     = (tid >> 4) & 1;  // half-wave selector
  const int w      = tid >> 5;        // wave id, 0..7

  const float* Gr = G + (2 - STAGE) * 512;  // gate real, row-major 16x16
  const float* Gi = Gr + 256;               // gate imag

  // A-operand registers: A[M=lane16][K = 4*s + 2*hi + {0,1}]
  v2f ar[4], ai[4], an[4];
#pragma unroll
  for (int s = 0; s < 4; ++s) {
    const int k0 = 4 * s + 2 * hi;
    ar[s].x = Gr[lane16 * 16 + k0];
    ar[s].y = Gr[lane16 * 16 + k0 + 1];
    ai[s].x = Gi[lane16 * 16 + k0];
    ai[s].y = Gi[lane16 * 16 + k0 + 1];
    an[s].x = -ai[s].x;               // f32 WMMA has no A-neg modifier
    an[s].y = -ai[s].y;
  }

  // 16 tiles total, 2 per wave. Each tile reads exactly the 256 LDS
  // elements it writes -> in-place within the stage is safe.
#pragma unroll
  for (int tt = 0; tt < 2; ++tt) {
    const int t = w * 2 + tt;

    // B-operand: B[K = 4*s + 2*hi + {0,1}][N = lane16]
    v2f br[4], bi[4];
#pragma unroll
    for (int s = 0; s < 4; ++s) {
      const int k0 = 4 * s + 2 * hi;
      br[s].x = SR[map_idx<STAGE>(t, k0,     lane16)];
      br[s].y = SR[map_idx<STAGE>(t, k0 + 1, lane16)];
      bi[s].x = SI[map_idx<STAGE>(t, k0,     lane16)];
      bi[s].y = SI[map_idx<STAGE>(t, k0 + 1, lane16)];
    }

    // Complex GEMM: accR = Ur*Xr - Ui*Xi ; accI = Ur*Xi + Ui*Xr
    v8f accR = {};
    v8f accI = {};
#pragma unroll
    for (int s = 0; s < 4; ++s) accR = wmma_f32(ar[s], br[s], accR);
#pragma unroll
    for (int s = 0; s < 4; ++s) accR = wmma_f32(an[s], bi[s], accR);
#pragma unroll
    for (int s = 0; s < 4; ++s) accI = wmma_f32(ar[s], bi[s], accI);
#pragma unroll
    for (int s = 0; s < 4; ++s) accI = wmma_f32(ai[s], br[s], accI);

    // D layout: element r holds [M = r + 8*hi][N = lane16]
#pragma unroll
    for (int r = 0; r < 8; ++r) {
      const int kout = r + 8 * hi;
      const int idx  = map_idx<STAGE>(t, kout, lane16);
      if constexpr (STAGE == 2) {   // final stage: straight to HBM
        gOutR[idx] = accR[r];
        gOutI[idx] = accI[r];
      } else {                      // intermediate: back into LDS in place
        SR[idx] = accR[r];
        SI[idx] = accI[r];
      }
    }
  }
}

__global__ __launch_bounds__(256) void ulayer_kernel(
    const float* __restrict__ thetas,  // [3]
    const float* __restrict__ evecs,   // [16,16] row-major, V
    const float* __restrict__ evals,   // [16]
    const float* __restrict__ stateR,  // [128,4096]
    const float* __restrict__ stateI,  // [128,4096]
    float* __restrict__ out) {         // [2,128,4096]
  __shared__ __align__(16) float SR[DIM];   // 16 KB
  __shared__ __align__(16) float SI[DIM];   // 16 KB
  __shared__ float G[3 * 512];              //  6 KB: [gate][re|im][16*16]

  const int tid = threadIdx.x;
  const int b   = blockIdx.x;

  // ---- Stage the state into LDS (b128 global loads -> b128 ds stores) ----
  const float4* gr4 = (const float4*)(stateR + b * DIM);
  const float4* gi4 = (const float4*)(stateI + b * DIM);
  float4* sr4 = (float4*)SR;
  float4* si4 = (float4*)SI;
#pragma unroll
  for (int i = 0; i < 4; ++i) {
    sr4[tid + i * 256] = gr4[tid + i * 256];
    si4[tid + i * 256] = gi4[tid + i * 256];
  }

  // ---- Build the 16x16 gates: U = V diag(exp(-i*theta*lambda)) V^T ----
  {
    const int r = tid >> 4, c = tid & 15;
    float vr[16], vc[16], lam[16];
#pragma unroll
    for (int k = 0; k < 16; ++k) {
      vr[k]  = evecs[r * 16 + k];
      vc[k]  = evecs[c * 16 + k];
      lam[k] = evals[k];
    }
#pragma unroll
    for (int g = 0; g < 3; ++g) {
      const float theta = thetas[g];
      float sre = 0.f, sim = 0.f;
#pragma unroll
      for (int k = 0; k < 16; ++k) {
        float sn, cs;
        __sincosf(theta * lam[k], &sn, &cs);
        const float vv = vr[k] * vc[k];
        sre += vv * cs;   // Re U[r][c]
        sim -= vv * sn;   // Im U[r][c]
      }
      G[g * 512 + tid]       = sre;
      G[g * 512 + 256 + tid] = sim;
    }
  }

  __syncthreads();

  do_stage<0>(G, SR, SI, nullptr, nullptr);   // contract c with U3
  __syncthreads();
  do_stage<1>(G, SR, SI, nullptr, nullptr);   // contract b with U2
  __syncthreads();
  do_stage<2>(G, SR, SI,                       // contract a with U1 -> HBM
              out + b * DIM,                   // real plane
              out + (BATCH + b) * DIM);        // imag plane
}

extern "C" void kernel_launch(void* const* d_in, const int* in_sizes, int n_in,
                              void* d_out, int out_size, void* d_ws, size_t ws_size,
                              hipStream_t stream) {
  (void)in_sizes; (void)n_in; (void)d_ws; (void)ws_size; (void)out_size;
  const float* thetas = (const float*)d_in[0];
  const float* evecs  = (const float*)d_in[1];
  const float* evals  = (const float*)d_in[2];
  const float* sreal  = (const float*)d_in[3];
  const float* simag  = (const float*)d_in[4];
  float* out = (float*)d_out;

  ulayer_kernel<<<BATCH, 256, 0, stream>>>(thetas, evecs, evals, sreal, simag, out);
}